// ClassifierJax_65893388255613
// MI455X (gfx1250) — compile-verified
//
#include <hip/hip_runtime.h>
#include <hip/hip_bf16.h>

typedef float v2f __attribute__((ext_vector_type(2)));
typedef float v8f __attribute__((ext_vector_type(8)));
typedef unsigned int v4u __attribute__((ext_vector_type(4)));
typedef int v4i __attribute__((ext_vector_type(4)));
typedef int v8i __attribute__((ext_vector_type(8)));

#define B_TOT 128
#define L_LEN 1024
#define INC 16
#define HID 64
#define OUTN 10
#define MAX_STEPS 128
#define G 16          // samples per workgroup (== WMMA N)
#define TPB 256       // 8 waves

// ---- LDS layout (float offsets) ----
#define OFF_W2   0        // [1024][64]  256 KB
#define OFF_W1   65536    // [64][64]    16 KB
#define OFF_B1   69632    // [64]
#define OFF_B2   69696    // [1024]
#define OFF_Y    70720    // [64][16]
#define OFF_YS   71744    // [64][16]
#define OFF_H1   72768    // [64][16]
#define OFF_DX   73792    // [16][16]
#define OFF_K    74048    // [7][64][16]
#define OFF_ST   81216    // t per sample
#define OFF_DT   81232    // dt
#define OFF_DTC  81248    // clamped dt
#define OFF_T1   81264    // ts[-1]
#define OFF_ES   81280    // err^2 sum
#define OFF_OK   81296    // active / accept flag
#define SMEM_FLOATS 81312 // 325,248 bytes < 320 KB WGP LDS

// gfx1250 hardware tanh (V_TANH_F32, TRANS pipe — co-executes with WMMA)
__device__ __forceinline__ float htanh(float x) { return __builtin_amdgcn_tanhf(x); }

// Dopri5 tableau (stage 6 row == b5 => YS at stage 6 IS y5)
__constant__ float CC[7] = {0.f, 0.2f, 0.3f, 0.8f, 8.f/9.f, 1.f, 1.f};
__constant__ float AT[7][6] = {
  {0,0,0,0,0,0},
  {0.2f,0,0,0,0,0},
  {3.f/40.f, 9.f/40.f,0,0,0,0},
  {44.f/45.f, -56.f/15.f, 32.f/9.f,0,0,0},
  {19372.f/6561.f, -25360.f/2187.f, 64448.f/6561.f, -212.f/729.f,0,0},
  {9017.f/3168.f, -355.f/33.f, 46732.f/5247.f, 49.f/176.f, -5103.f/18656.f,0},
  {35.f/384.f, 0.f, 500.f/1113.f, 125.f/192.f, -2187.f/6784.f, 11.f/84.f}
};
__constant__ float EE[7] = {
  35.f/384.f - 5179.f/57600.f, 0.f,
  500.f/1113.f - 7571.f/16695.f,
  125.f/192.f - 393.f/640.f,
  -2187.f/6784.f + 92097.f/339200.f,
  11.f/84.f - 187.f/2100.f,
  -1.f/40.f
};

// Cubic-Hermite (Catmull-Rom tangent) spline derivative at scalar t, channel c
__device__ __forceinline__ float spline_dx(const float* __restrict__ tsr,
                                           const float* __restrict__ xsr,
                                           float t, int c) {
  int lo = 0, hi = L_LEN;
  while (lo < hi) { int mid = (lo + hi) >> 1; if (tsr[mid] <= t) lo = mid + 1; else hi = mid; }
  int i = lo - 1; if (i < 0) i = 0; if (i > L_LEN - 2) i = L_LEN - 2;
  float t0 = tsr[i], t1 = tsr[i + 1];
  float h = t1 - t0;
  float s = (t - t0) / h;
  float xi  = xsr[i * INC + c];
  float xi1 = xsr[(i + 1) * INC + c];
  float mi, mi1;
  if (i >= 1) mi = (xi1 - xsr[(i - 1) * INC + c]) / (t1 - tsr[i - 1]);
  else        mi = (xsr[INC + c] - xsr[c]) / (tsr[1] - tsr[0]);
  if (i + 1 <= L_LEN - 2) mi1 = (xsr[(i + 2) * INC + c] - xi) / (tsr[i + 2] - t0);
  else mi1 = (xsr[(L_LEN - 1) * INC + c] - xsr[(L_LEN - 2) * INC + c]) /
             (tsr[L_LEN - 1] - tsr[L_LEN - 2]);
  float s2 = s * s;
  return ((6.f * s2 - 6.f * s) * (xi - xi1)) / h +
         (3.f * s2 - 4.f * s + 1.f) * mi + (3.f * s2 - 2.f * s) * mi1;
}

__global__ __launch_bounds__(TPB) void cde_wmma_kernel(
    const float* __restrict__ ts, const float* __restrict__ xs,
    const float* __restrict__ W1, const float* __restrict__ b1,
    const float* __restrict__ W2, const float* __restrict__ b2,
    const float* __restrict__ lin_w, const float* __restrict__ lin_b,
    float* __restrict__ out) {
  extern __shared__ float sm[];
  float* sW2  = sm + OFF_W2;
  float* sW1  = sm + OFF_W1;
  float* sB1  = sm + OFF_B1;
  float* sB2  = sm + OFF_B2;
  float* sY   = sm + OFF_Y;
  float* sYS  = sm + OFF_YS;
  float* sH1  = sm + OFF_H1;
  float* sDX  = sm + OFF_DX;
  float* sK   = sm + OFF_K;
  float* sT   = sm + OFF_ST;
  float* sDT  = sm + OFF_DT;
  float* sDTC = sm + OFF_DTC;
  float* sT1  = sm + OFF_T1;
  float* sES  = sm + OFF_ES;
  float* sOK  = sm + OFF_OK;

  const int tid  = threadIdx.x;
  const int lane = tid & 31;
  const int wv   = tid >> 5;
  const int lm   = lane & 15;    // WMMA N / row-in-tile index
  const int hih  = lane >> 4;    // lane-half
  const int grp  = blockIdx.x;
  const int bsmp = tid >> 4;     // sample for the spline phase
  const int chan = tid & 15;     // channel for the spline phase

  // ---- W2 (256 KB) via Tensor Data Mover: 2-D tensor 1024x64 f32, whole-tile ----
  if (tid < 32) {
    unsigned long long ga = (unsigned long long)(uintptr_t)W2;
    unsigned int ldsa = (unsigned int)(uintptr_t)sW2;
    v4u g0v = { 1u,                                    // count=1, gather off
                ldsa,                                  // lds_addr
                (unsigned int)ga,                      // global_addr[31:0]
                (unsigned int)((ga >> 32) & 0x1FFFFFFull) | (2u << 30) }; // [56:32] | type=2
    v8i g1v = { 0x20000,                 // data_size=4B, no mask/pad/iterate
                (int)(64u << 16),        // tensor_dim0 = 64
                (int)(1024u << 16),      // tensor_dim1 = 1024
                (int)(64u << 16),        // tile_dim0   = 64
                1024,                    // tile_dim1   = 1024, tile_dim2 = 0
                64,                      // tensor_dim0_stride = 64
                0, 0 };
    v4i zz4 = {0, 0, 0, 0};
    v8i zz8 = {0, 0, 0, 0, 0, 0, 0, 0};
    __builtin_amdgcn_tensor_load_to_lds(g0v, g1v, zz4, zz4, zz8, 0);
    __builtin_amdgcn_s_wait_tensorcnt(0);
  }
  // small weights cooperatively
  for (int i = tid; i < HID * HID; i += TPB) sW1[i] = W1[i];
  for (int i = tid; i < HID * INC; i += TPB) sB2[i] = b2[i];
  if (tid < HID) sB1[tid] = b1[tid];
  // state init
  for (int q = 0; q < 4; ++q) { int e = tid * 4 + q; sY[e] = 0.f; sYS[e] = 0.f; }
  if (tid < G) {
    const float* tsr = ts + (grp * G + tid) * L_LEN;
    sT[tid]  = tsr[0];
    sT1[tid] = tsr[L_LEN - 1];
    sDT[tid] = 0.01f;
    sDTC[tid] = 0.01f; sES[tid] = 0.f; sOK[tid] = 0.f;
  }
  __syncthreads();

  const float* tsrow = ts + (grp * G + bsmp) * L_LEN;
  const float* xsrow = xs + (size_t)(grp * G + bsmp) * L_LEN * INC;

  for (int step = 0; step < MAX_STEPS; ++step) {
    if (tid < G) {
      float rem = sT1[tid] - sT[tid];
      int active = rem > 1e-12f;
      float dt = sDT[tid];
      sDTC[tid] = active ? fminf(dt, rem) : dt;
      sOK[tid] = active ? 1.f : 0.f;   // temporarily: "active"
      sES[tid] = 0.f;
    }
    __syncthreads();

    for (int stg = 0; stg < 7; ++stg) {
      // ---- stage input YS = y + dt_c * sum_j a[stg][j] * k_j   (stage 6 => y5) ----
      {
        int e0 = tid * 4;
#pragma unroll
        for (int q = 0; q < 4; ++q) {
          int e = e0 + q; int b = e & 15;
          float acc = sY[e];
          float dtc = sDTC[b];
#pragma unroll
          for (int j = 0; j < 6; ++j)
            if (j < stg) acc += dtc * AT[stg][j] * sK[j * 1024 + e];
          sYS[e] = acc;
        }
      }
      // ---- spline derivative dx[c] at t_eval, per (sample, channel) thread ----
      {
        float tev = sT[bsmp] + CC[stg] * sDTC[bsmp];
        sDX[chan * G + bsmp] = spline_dx(tsrow, xsrow, tev, chan);
      }
      __syncthreads();

      // ---- GEMM1: H1 = tanh(W1 @ YS + b1), [64x16], waves 0..3 one tile each ----
      if (wv < 4) {
        v8f acc = {0.f,0.f,0.f,0.f,0.f,0.f,0.f,0.f};
        const float* arow = sW1 + (16 * wv + lm) * HID + 2 * hih;
        const float* bcol = sYS + 2 * hih * G + lm;
#pragma unroll
        for (int kk = 0; kk < 16; ++kk) {
          v2f af = *(const v2f*)(arow + 4 * kk);          // ds_load_b64
          v2f bf; bf.x = bcol[(4 * kk) * G]; bf.y = bcol[(4 * kk + 1) * G];
          acc = __builtin_amdgcn_wmma_f32_16x16x4_f32(false, af, false, bf,
                                                      (short)0, acc, false, false);
        }
#pragma unroll
        for (int v = 0; v < 8; ++v) {
          int hr = 16 * wv + v + 8 * hih;
          sH1[hr * G + lm] = htanh(acc[v] + sB1[hr]);
        }
      }
      __syncthreads();

      // ---- GEMM2: F = tanh(W2 @ H1 + b2) with 8 live accumulators per wave.
      //      k-chunk outer loop, fully unrolled: B-fragment loaded ONCE per
      //      chunk and reused by 8 independent WMMAs (8 M-tiles); scheduler can
      //      hoist LDS loads across chunks instead of per-WMMA dscnt stalls. ----
      {
        v8f acc[8];
#pragma unroll
        for (int jt = 0; jt < 8; ++jt)
          acc[jt] = (v8f){0.f,0.f,0.f,0.f,0.f,0.f,0.f,0.f};
        const float* a0   = sW2 + (16 * (8 * wv) + lm) * HID + 2 * hih;
        const float* bcol = sH1 + 2 * hih * G + lm;
#pragma unroll
        for (int kk = 0; kk < 16; ++kk) {
          v2f bf; bf.x = bcol[(4 * kk) * G]; bf.y = bcol[(4 * kk + 1) * G];
          v2f af[8];
#pragma unroll
          for (int jt = 0; jt < 8; ++jt)
            af[jt] = *(const v2f*)(a0 + jt * 16 * HID + 4 * kk); // ds_load_b64
#pragma unroll
          for (int jt = 0; jt < 8; ++jt)
            acc[jt] = __builtin_amdgcn_wmma_f32_16x16x4_f32(false, af[jt], false, bf,
                                                            (short)0, acc[jt], false, false);
        }
        // epilogue: tanh(+b2), contract with dx over channel dim, fold halves
#pragma unroll
        for (int jt = 0; jt < 8; ++jt) {
          int T = 8 * wv + jt;           // output h index (tile == h)
          float part = 0.f;
#pragma unroll
          for (int v = 0; v < 8; ++v) {
            int c = v + 8 * hih;
            float f = htanh(acc[jt][v] + sB2[16 * T + c]);
            part += f * sDX[c * G + lm];
          }
          part += __shfl_xor(part, 16, 32);  // full sum over c
          if (lane < 16) sK[stg * 1024 + T * G + lm] = part;
        }
      }
      __syncthreads();
    } // stages

    // ---- error norm, accept/step-size control, masked commit ----
    {
      int e0 = tid * 4;
#pragma unroll
      for (int q = 0; q < 4; ++q) {
        int e = e0 + q; int b = e & 15;
        float err = 0.f;
#pragma unroll
        for (int j = 0; j < 7; ++j) err += EE[j] * sK[j * 1024 + e];
        err *= sDTC[b];
        float ynew = sYS[e];   // y5 (stage-6 combo)
        float scale = 1e-3f + 1e-3f * fmaxf(fabsf(sY[e]), fabsf(ynew));
        float r = err / scale;
        atomicAdd(&sES[b], r * r);
      }
    }
    __syncthreads();
    if (tid < G) {
      float en = sqrtf(sES[tid] * (1.f / 64.f));
      int active = sOK[tid] > 0.5f;
      int accept = en <= 1.f;
      float factor = 0.9f * powf(en + 1e-10f, -0.2f);
      factor = fminf(fmaxf(factor, 0.2f), 5.f);
      int ok = active && accept;
      if (ok) sT[tid] = sT[tid] + sDTC[tid];
      if (active) sDT[tid] = sDTC[tid] * factor;
      sOK[tid] = ok ? 1.f : 0.f;
    }
    __syncthreads();
    {
      int e0 = tid * 4;
#pragma unroll
      for (int q = 0; q < 4; ++q) {
        int e = e0 + q; int b = e & 15;
        if (sOK[b] > 0.5f) sY[e] = sYS[e];
      }
    }
    __syncthreads();
  } // steps

  // ---- classifier head: out[b,o] = y_final . lin_w[o] + lin_b[o] ----
  if (tid < G * OUTN) {
    int b = tid / OUTN, o = tid % OUTN;
    float acc = lin_b[o];
    for (int h = 0; h < HID; ++h) acc += sY[h * G + b] * lin_w[o * HID + h];
    out[(grp * G + b) * OUTN + o] = acc;
  }
}

extern "C" void kernel_launch(void* const* d_in, const int* in_sizes, int n_in,
                              void* d_out, int out_size, void* d_ws, size_t ws_size,
                              hipStream_t stream) {
  (void)in_sizes; (void)n_in; (void)d_ws; (void)ws_size; (void)out_size;
  const float* ts  = (const float*)d_in[0];
  const float* xs  = (const float*)d_in[1];
  const float* W1  = (const float*)d_in[2];
  const float* b1  = (const float*)d_in[3];
  const float* W2  = (const float*)d_in[4];
  const float* b2  = (const float*)d_in[5];
  const float* lw  = (const float*)d_in[6];
  const float* lb  = (const float*)d_in[7];
  float* out = (float*)d_out;
  size_t shmem = (size_t)SMEM_FLOATS * sizeof(float);
  (void)hipFuncSetAttribute(reinterpret_cast<const void*>(cde_wmma_kernel),
                            hipFuncAttributeMaxDynamicSharedMemorySize, (int)shmem);
  cde_wmma_kernel<<<B_TOT / G, TPB, shmem, stream>>>(ts, xs, W1, b1, W2, b2, lw, lb, out);
}